// RLSE_53231824667231
// MI455X (gfx1250) — compile-verified
//
#include <hip/hip_runtime.h>

// CDNA5 (gfx1250) wave32 WMMA types
typedef __attribute__((ext_vector_type(2))) float v2f;  // 16x4 / 4x16 f32 fragment (2 VGPRs)
typedef __attribute__((ext_vector_type(8))) float v8f;  // 16x16 f32 C/D tile (8 VGPRs)

#define NVAR 128
#define KBLK 16            // rows folded per Woodbury block step
#define NBLK 512           // 8192 / 16 sequential block steps
#define SST  132           // padded LDS stride for S (128 + 4)
#define MST  36            // padded stride for augmented [M | I] (32 + 4)

// LDS layout (floats)
#define LDS_FLOATS (NVAR*SST + 2*KBLK*NVAR + 2*KBLK*NVAR + 16*MST + NVAR + 16 + 32)

static __device__ __forceinline__ v8f wmma_f32_4(v2f a, v2f b, v8f c) {
  return __builtin_amdgcn_wmma_f32_16x16x4_f32(false, a, false, b, (short)0, c,
                                               false, false);
}

// uniform-lane broadcast: v_readlane_b32 (k is wave-uniform, unrolled -> immediate)
static __device__ __forceinline__ float bcast(float v, int l) {
  return __int_as_float(__builtin_amdgcn_readlane(__float_as_int(v), l));
}

// A-style fragment: rows = M (lane l16), K pair selected by half
static __device__ __forceinline__ v2f ldA(const float* m, int stride, int kb,
                                          int l16, int half) {
  v2f f; f.x = m[l16*stride + kb + 2*half];
         f.y = m[l16*stride + kb + 2*half + 1];
  return f;
}
// B-style fragment: rows = K (pair by half), cols = N (lane l16)
static __device__ __forceinline__ v2f ldB(const float* m, int stride, int kb,
                                          int cb, int l16, int half) {
  const int r0 = kb + 2*half;
  v2f f; f.x = m[r0*stride + cb + l16];
         f.y = m[(r0+1)*stride + cb + l16];
  return f;
}

extern "C" __global__ __launch_bounds__(256)
void rls_block_wmma(const float* __restrict__ gA,
                    const float* __restrict__ gb,
                    const float* __restrict__ gS0,
                    const float* __restrict__ gTh0,
                    float* __restrict__ out)
{
  extern __shared__ float lds[];
  float* lS  = lds;
  float* lA  = lS  + NVAR*SST;       // 2 buffers of 16x128
  float* lP  = lA  + 2*KBLK*NVAR;
  float* lU  = lP  + KBLK*NVAR;
  float* lM  = lU  + KBLK*NVAR;
  float* lth = lM  + 16*MST;
  float* le  = lth + NVAR;
  float* lb  = le  + 16;             // 2 buffers of 16

  const int tid  = threadIdx.x;
  const int lane = tid & 31;
  const int wv   = tid >> 5;    // wave id 0..7
  const int half = lane >> 4;   // half-wave selects K pair / M half (ISA layout)
  const int l16  = lane & 15;

  // ---- initialize state from s0 / theta0; stage block 0 ----
  for (int i = tid; i < NVAR*NVAR; i += 256)
    lS[(i >> 7) * SST + (i & 127)] = gS0[i];
  if (tid < NVAR) lth[tid] = gTh0[tid];
  for (int i = tid; i < (KBLK*NVAR)/4; i += 256)
    ((float4*)lA)[i] = ((const float4*)gA)[i];
  if (tid < KBLK) lb[tid] = gb[tid];
  __syncthreads();

  for (int blk = 0; blk < NBLK; ++blk) {
    const int cur = blk & 1;
    const float* lAc = lA + cur * KBLK*NVAR;         // current block
    float*       lAn = lA + (cur ^ 1) * KBLK*NVAR;   // next block buffer
    const float* lbc = lb + cur * KBLK;
    float*       lbn = lb + (cur ^ 1) * KBLK;

    // ---- P = A_blk (16x128) * S (128x128); wave wv owns cols [16wv,16wv+16) ----
    // software-pipelined: issue chunk k+1 fragment loads before chunk k WMMA
    {
      const int cb = wv * 16;
      v8f acc = {};
      v2f af = ldA(lAc, NVAR, 0, l16, half);
      v2f bf = ldB(lS, SST, 0, cb, l16, half);
      #pragma unroll
      for (int kc = 0; kc < 32; ++kc) {
        v2f afn = af, bfn = bf;
        if (kc + 1 < 32) {
          afn = ldA(lAc, NVAR, (kc+1)*4, l16, half);
          bfn = ldB(lS, SST, (kc+1)*4, cb, l16, half);
        }
        acc = wmma_f32_4(af, bf, acc);
        af = afn; bf = bfn;
      }
      for (int r = 0; r < 8; ++r)
        lP[(8*half + r)*NVAR + cb + l16] = acc[r];
    }
    __syncthreads();

    if (wv == 0) {
      // ---- wave 0: M = I + P*A^T (WMMA), then register-resident Gauss-Jordan ----
      v8f g = {};
      v2f af = ldA(lP, NVAR, 0, l16, half);
      v2f bf = ldA(lAc, NVAR, 0, l16, half);   // B = A^T -> A-style load
      #pragma unroll
      for (int kc = 0; kc < 32; ++kc) {
        v2f afn = af, bfn = bf;
        if (kc + 1 < 32) {
          afn = ldA(lP, NVAR, (kc+1)*4, l16, half);
          bfn = ldA(lAc, NVAR, (kc+1)*4, l16, half);
        }
        g = wmma_f32_4(af, bf, g);
        af = afn; bf = bfn;
      }
      // stage augmented [M+I | I] tile (wave-internal LDS ordering, no barrier)
      for (int r = 0; r < 8; ++r) {
        const int m = 8*half + r;
        lM[m*MST + l16]      = g[r] + ((m == l16) ? 1.f : 0.f);
        lM[m*MST + 16 + l16] = (m == l16) ? 1.f : 0.f;
      }
      // lane = augmented column (0..31); 16 row registers per lane
      float row[16];
      #pragma unroll
      for (int r = 0; r < 16; ++r) row[r] = lM[r*MST + lane];
      // pivot-free GJ (M is SPD): per pivot, readlane broadcasts of column k
      #pragma unroll
      for (int k = 0; k < 16; ++k) {
        const float ip = 1.f / bcast(row[k], k);
        row[k] *= ip;
        #pragma unroll
        for (int r = 0; r < 16; ++r) {
          if (r == k) continue;
          const float fr = bcast(row[r], k);   // pre-update col-k entry of row r
          row[r] = fmaf(-fr, row[k], row[r]);
        }
      }
      // write Minv (augmented cols 16..31) back for all waves
      if (lane >= 16) {
        #pragma unroll
        for (int r = 0; r < 16; ++r) lM[r*MST + lane] = row[r];
      }
    } else {
      // ---- waves 1..7: overlap — stage next A/b block; wave 1 computes e ----
      const int t2 = tid - 32;  // 0..223
      if (blk + 1 < NBLK) {
        const float4* Abn = (const float4*)(gA + (size_t)(blk + 1) * KBLK*NVAR);
        for (int i = t2; i < (KBLK*NVAR)/4; i += 224) ((float4*)lAn)[i] = Abn[i];
        if (t2 < KBLK) lbn[t2] = gb[(blk + 1) * KBLK + t2];
        if (blk + 2 < NBLK)
          __builtin_prefetch(gA + (size_t)(blk + 2) * KBLK*NVAR + t2 * 8, 0, 0);
      }
      if (wv == 1 && lane < KBLK) {  // innovation e = b - A*theta (old theta)
        float s = 0.f;
        for (int j = 0; j < NVAR; ++j) s = fmaf(lAc[lane*NVAR + j], lth[j], s);
        le[lane] = lbc[lane] - s;
      }
    }
    __syncthreads();

    // ---- U = Minv (16x16) * P (16x128); K = 16 -> 4 pipelined WMMA chunks ----
    {
      const int cb = wv * 16;
      v8f acc = {};
      v2f af = ldA(lM + 16, MST, 0, l16, half);  // Minv lives in cols 16..31
      v2f bf = ldB(lP, NVAR, 0, cb, l16, half);
      #pragma unroll
      for (int kc = 0; kc < 4; ++kc) {
        v2f afn = af, bfn = bf;
        if (kc + 1 < 4) {
          afn = ldA(lM + 16, MST, (kc+1)*4, l16, half);
          bfn = ldB(lP, NVAR, (kc+1)*4, cb, l16, half);
        }
        acc = wmma_f32_4(af, bf, acc);
        af = afn; bf = bfn;
      }
      for (int r = 0; r < 8; ++r)
        lU[(8*half + r)*NVAR + cb + l16] = acc[r];
    }
    __syncthreads();

    // ---- S -= P^T * U : wave wv owns S row-block [16wv,16wv+16), 8 col tiles ----
    {
      const int rb = wv * 16;
      for (int tj = 0; tj < 8; ++tj) {
        const int cb2 = tj * 16;
        v8f c;
        for (int r = 0; r < 8; ++r)
          c[r] = lS[(rb + 8*half + r)*SST + cb2 + l16];
        // A = -P^T : f32 WMMA NEG only covers C -> negate fragment in VALU.
        // P^T fragment is a B-style load from P at column block rb.
        v2f af = ldB(lP, NVAR, 0, rb, l16, half);
        v2f bf = ldB(lU, NVAR, 0, cb2, l16, half);
        #pragma unroll
        for (int kc = 0; kc < 4; ++kc) {
          v2f afn = af, bfn = bf;
          if (kc + 1 < 4) {
            afn = ldB(lP, NVAR, (kc+1)*4, rb, l16, half);
            bfn = ldB(lU, NVAR, (kc+1)*4, cb2, l16, half);
          }
          v2f an; an.x = -af.x; an.y = -af.y;
          c = wmma_f32_4(an, bf, c);
          af = afn; bf = bfn;
        }
        for (int r = 0; r < 8; ++r)
          lS[(rb + 8*half + r)*SST + cb2 + l16] = c[r];
      }
    }

    // ---- theta += U^T * e  (gain K = S_new A^T = (Minv P)^T = U^T) ----
    if (tid < NVAR) {
      float s = 0.f;
      for (int r = 0; r < KBLK; ++r) s = fmaf(lU[r*NVAR + tid], le[r], s);
      lth[tid] += s;
    }
    __syncthreads();
  }

  if (tid < NVAR) out[tid] = lth[tid];
}

extern "C" void kernel_launch(void* const* d_in, const int* in_sizes, int n_in,
                              void* d_out, int out_size, void* d_ws, size_t ws_size,
                              hipStream_t stream) {
  const float* gA   = (const float*)d_in[0];  // (8,1024,128) f32
  const float* gb   = (const float*)d_in[1];  // (8,1024)     f32
  const float* gS0  = (const float*)d_in[2];  // (128,128)    f32
  const float* gTh0 = (const float*)d_in[3];  // (128,1)      f32
  float* out = (float*)d_out;                 // (128,1)      f32

  const size_t ldsBytes = (size_t)LDS_FLOATS * sizeof(float);
  rls_block_wmma<<<1, 256, ldsBytes, stream>>>(gA, gb, gS0, gTh0, out);
}